// PEwithPeak_69827578298900
// MI455X (gfx1250) — compile-verified
//
#include <hip/hip_runtime.h>
#include <hip/hip_bf16.h>
#include <stdint.h>

typedef __attribute__((ext_vector_type(4))) float v4f;

#define S_LEN 4096
#define BATCH 64
#define DIM   256
#define NPEAK 8
#define D4    (DIM / 4)           // 64 float4 per row
#define PLANE4 (BATCH * D4)       // 4096 float4 per sequence position

// ---------------------------------------------------------------------------
// Kernel 1: out[s,b,:] = x[s,b,:] + pe[s,:]
// One block per sequence position s. pe row (1 KB) is staged into LDS with the
// gfx1250 async global->LDS data mover (ASYNCcnt path), then reused by all 64
// batches handled by this block. x/out are streamed with non-temporal b128
// accesses so the 512 MB stream does not thrash L2.
// ---------------------------------------------------------------------------
__global__ __launch_bounds__(256) void pe_add_kernel(const float* __restrict__ x,
                                                     const float* __restrict__ pe,
                                                     float* __restrict__ out) {
    __shared__ v4f spe[D4];

    const int s   = blockIdx.x;
    const int tid = threadIdx.x;

    // Stage pe[s, 0:256] (64 x b128) into LDS via async tensor-class copy.
    // tid<64 is wave-uniform under wave32 (waves 0 and 1 fully active).
    if (tid < D4) {
        const v4f* src = (const v4f*)pe + (size_t)s * D4 + tid;
        unsigned lds_addr = (unsigned)(uintptr_t)(&spe[tid]); // low 32b of flat = LDS byte offset
        asm volatile("global_load_async_to_lds_b128 %0, %1, off"
                     :
                     : "v"(lds_addr), "v"(src)
                     : "memory");
    }
    // Drain ASYNCcnt (no-op for waves that issued nothing), then make the LDS
    // tile visible to all 8 waves of the block.
    asm volatile("s_wait_asynccnt 0x0" ::: "memory");
    __syncthreads();

    // Each thread owns one d4 column (tid & 63) -> single pe vector for all
    // 16 iterations; walk b in steps of 4.
    const v4f pv = spe[tid & (D4 - 1)];

    const v4f* __restrict__ xin = (const v4f*)x   + (size_t)s * PLANE4 + tid;
    v4f* __restrict__       o   = (v4f*)out       + (size_t)s * PLANE4 + tid;

#pragma unroll
    for (int i = 0; i < 16; ++i) {
        v4f v = __builtin_nontemporal_load(xin + i * 256);
        v += pv;
        __builtin_nontemporal_store(v, o + i * 256);
    }
}

// ---------------------------------------------------------------------------
// Kernel 2: deterministic scatter-add of peak embeddings.
// One block per batch b; its 8 peaks are processed sequentially, so duplicate
// positions accumulate in a fixed order (bit-deterministic, no atomics).
// 64 threads x float4 = one 256-float row per step.
// ---------------------------------------------------------------------------
__global__ __launch_bounds__(64) void peak_scatter_kernel(const int* __restrict__ peaks,
                                                          const float* __restrict__ table,
                                                          float* __restrict__ out) {
    const int b = blockIdx.x;   // 0..63
    const int t = threadIdx.x;  // 0..63

    const v4f* __restrict__ tbl = (const v4f*)table;
    v4f* __restrict__ o = (v4f*)out;

#pragma unroll
    for (int p = 0; p < NPEAK; ++p) {
        const int pos = peaks[b * NPEAK + p];
        if (pos >= 0 && pos < S_LEN) {
            const v4f e = tbl[(size_t)pos * D4 + t];
            const size_t oi = ((size_t)pos * BATCH + b) * D4 + t;
            v4f cur = o[oi];
            cur += e;
            o[oi] = cur;   // same thread handles any duplicate pos -> ordered RMW
        }
    }
}

extern "C" void kernel_launch(void* const* d_in, const int* in_sizes, int n_in,
                              void* d_out, int out_size, void* d_ws, size_t ws_size,
                              hipStream_t stream) {
    const float* x     = (const float*)d_in[0];  // [S,B,D] fp32
    const int*   peaks = (const int*)  d_in[1];  // [B,P] int32
    const float* pe    = (const float*)d_in[2];  // [MAX_LEN,1,D] fp32
    const float* table = (const float*)d_in[3];  // [MAX_LEN,D] fp32
    float* out = (float*)d_out;

    (void)in_sizes; (void)n_in; (void)out_size; (void)d_ws; (void)ws_size;

    // Streaming add: one block per sequence position.
    pe_add_kernel<<<S_LEN, 256, 0, stream>>>(x, pe, out);
    // Deterministic per-batch scatter (runs after kernel 1 on the same stream).
    peak_scatter_kernel<<<BATCH, 64, 0, stream>>>(peaks, table, out);
}